// pointnet2_seg_msg_radar_42906723287435
// MI455X (gfx1250) — compile-verified
//
#include <hip/hip_runtime.h>
#include <cstdint>

// ---------------------------------------------------------------------------
// PointNet++ MSG segmentation for MI455X (gfx1250).
// Dense shared-MLP GEMMs run on V_WMMA_F32_16X16X4_F32 (f32 matrix path,
// wave32). Each wave owns a 32x32 output block (2x2 WMMA tiles) and the
// K-dim is stride-padded to a multiple of 4 so the inner loop is pure
// global_load_b64 + v_wmma with no predication (EXEC must stay all-1s).
// ---------------------------------------------------------------------------

typedef __attribute__((ext_vector_type(2))) float v2f;
typedef __attribute__((ext_vector_type(8))) float v8f;

#define TPB 256
#define BATCH 8

// ---------------- Farthest point sampling: one block per batch -------------
__global__ __launch_bounds__(1024)
void fps_kernel(const float* __restrict__ xyz, int N, int M, int* __restrict__ out_idx)
{
    __shared__ float dist[4096];
    __shared__ float rd[1024];
    __shared__ int   ri[1024];
    __shared__ int   cur;
    const int b   = blockIdx.x;
    const int tid = threadIdx.x;
    const int bs  = blockDim.x;
    const float* x = xyz + (size_t)b * N * 3;

    for (int i = tid; i < N; i += bs) dist[i] = 1e10f;
    if (tid == 0) cur = 0;
    __syncthreads();

    for (int it = 0; it < M; ++it) {
        const int c = cur;
        if (tid == 0) out_idx[b * M + it] = c;
        const float cx = x[c * 3 + 0], cy = x[c * 3 + 1], cz = x[c * 3 + 2];
        float bd = -1.0f; int bi = 0;
        for (int i = tid; i < N; i += bs) {
            float dx = x[i * 3 + 0] - cx;
            float dy = x[i * 3 + 1] - cy;
            float dz = x[i * 3 + 2] - cz;
            float d  = dx * dx + dy * dy + dz * dz;
            float nd = fminf(dist[i], d);
            dist[i] = nd;
            if (nd > bd) { bd = nd; bi = i; }   // strict > keeps first max
        }
        rd[tid] = bd; ri[tid] = bi;
        __syncthreads();
        for (int s = bs >> 1; s > 0; s >>= 1) {
            if (tid < s) {
                if (rd[tid + s] > rd[tid] ||
                    (rd[tid + s] == rd[tid] && ri[tid + s] < ri[tid])) {
                    rd[tid] = rd[tid + s]; ri[tid] = ri[tid + s];
                }
            }
            __syncthreads();
        }
        if (tid == 0) cur = ri[0];
        __syncthreads();
    }
}

// ---------------- gather new_xyz = xyz[fps_idx] ----------------------------
__global__ void gather_xyz_kernel(const float* __restrict__ xyz, const int* __restrict__ idx,
                                  int N, int M, float* __restrict__ out)
{
    int t = blockIdx.x * blockDim.x + threadIdx.x;
    if (t >= BATCH * M * 3) return;
    int c = t % 3; int q = t / 3; int b = q / M;
    out[t] = xyz[((size_t)b * N + idx[q]) * 3 + c];
}

// ---------------- ball query: first K indices (ascending) within radius ----
__global__ void ball_query_kernel(const float* __restrict__ xyz, const float* __restrict__ nxyz,
                                  int N, int M, float r2, int K, int* __restrict__ gi)
{
    int t = blockIdx.x * blockDim.x + threadIdx.x;
    if (t >= BATCH * M) return;
    int b = t / M;
    const float* x = xyz + (size_t)b * N * 3;
    const float qx = nxyz[t * 3 + 0], qy = nxyz[t * 3 + 1], qz = nxyz[t * 3 + 2];
    int* out = gi + (size_t)t * K;
    int cnt = 0, first = 0;
    bool found = false;
    for (int n = 0; n < N && cnt < K; ++n) {
        float dx = x[n * 3 + 0] - qx;
        float dy = x[n * 3 + 1] - qy;
        float dz = x[n * 3 + 2] - qz;
        float d  = dx * dx + dy * dy + dz * dz;
        if (d <= r2) {
            if (!found) { found = true; first = n; }
            out[cnt++] = n;
        }
    }
    for (; cnt < K; ++cnt) out[cnt] = first;
}

// ------ grouping into K-padded rows: feat[q,k,:] = [xyz[g]-nxyz, pts[g], 0] -
__global__ void group_kernel(const float* __restrict__ xyz, const float* __restrict__ pts,
                             const float* __restrict__ nxyz, const int* __restrict__ gi,
                             int N, int M, int K, int Cp, int Ctp, float* __restrict__ feat)
{
    const int Ct = Cp + 3;
    long t = (long)blockIdx.x * blockDim.x + threadIdx.x;
    long total = (long)BATCH * M * K * Ctp;
    if (t >= total) return;
    int  c = (int)(t % Ctp);
    long r = t / Ctp;
    int  k = (int)(r % K);
    long q = r / K;            // q = b*M + m
    long b = q / M;
    float v = 0.0f;
    if (c < Ct) {
        int g = gi[q * K + k];
        if (c < 3) v = xyz[((size_t)b * N + g) * 3 + c] - nxyz[q * 3 + c];
        else       v = pts[((size_t)b * N + g) * Cp + (c - 3)];
    }
    feat[t] = v;
}

// ---------------- zero-pad weight rows to stride Cinp ----------------------
__global__ void pad_w_kernel(const float* __restrict__ src, float* __restrict__ dst,
                             int Cout, int Cin, int Cinp)
{
    long t = (long)blockIdx.x * blockDim.x + threadIdx.x;
    long total = (long)Cout * Cinp;
    if (t >= total) return;
    int c = (int)(t % Cinp);
    int o = (int)(t / Cinp);
    dst[t] = (c < Cin) ? src[(size_t)o * Cin + c] : 0.0f;
}

// ---------------- WMMA f32 GEMM: D[P,Cout] = A[P,*] * W[Cout,*]^T ----------
// Requirements (guaranteed by the host code): P % 128 == 0, Cout % 32 == 0,
// Cinp % 4 == 0, A rows have stride Cinp (zero-padded), W rows stride Cinp.
// Block = 4 waves; each wave computes one 32x32 block (2x2 16x16 WMMA tiles).
__global__ __launch_bounds__(128)
void wmma_gemm_kernel(const float* __restrict__ A, const float* __restrict__ W,
                      float* __restrict__ D, int P, int Cinp, int Cout)
{
    const int wave = threadIdx.x >> 5;
    const int lane = threadIdx.x & 31;
    const int half = lane >> 4;      // 0: K=kb+0,1   1: K=kb+2,3
    const int l    = lane & 15;

    const int mb = blockIdx.x * 128 + wave * 32;
    const int nb = blockIdx.y * 32;

    v8f acc00 = {0.f,0.f,0.f,0.f,0.f,0.f,0.f,0.f};
    v8f acc01 = acc00, acc10 = acc00, acc11 = acc00;

    const float* Ar0 = A + (size_t)(mb + l)      * Cinp;
    const float* Ar1 = A + (size_t)(mb + 16 + l) * Cinp;
    const float* Wr0 = W + (size_t)(nb + l)      * Cinp;
    const float* Wr1 = W + (size_t)(nb + 16 + l) * Cinp;

    for (int kb = 0; kb < Cinp; kb += 4) {
        const int k0 = kb + half * 2;            // 8B-aligned offset
        v2f a0 = *(const v2f*)(Ar0 + k0);
        v2f a1 = *(const v2f*)(Ar1 + k0);
        v2f w0 = *(const v2f*)(Wr0 + k0);
        v2f w1 = *(const v2f*)(Wr1 + k0);
        acc00 = __builtin_amdgcn_wmma_f32_16x16x4_f32(false, a0, false, w0, (short)0, acc00, false, false);
        acc01 = __builtin_amdgcn_wmma_f32_16x16x4_f32(false, a0, false, w1, (short)0, acc01, false, false);
        acc10 = __builtin_amdgcn_wmma_f32_16x16x4_f32(false, a1, false, w0, (short)0, acc10, false, false);
        acc11 = __builtin_amdgcn_wmma_f32_16x16x4_f32(false, a1, false, w1, (short)0, acc11, false, false);
    }

    // store: tile (ti,tj): row = mb + ti*16 + r + half*8, col = nb + tj*16 + l
    const int m0 = mb + half * 8;
#pragma unroll
    for (int r = 0; r < 8; ++r) {
        float* d0 = D + (size_t)(m0 + r)      * Cout + nb;
        float* d1 = D + (size_t)(m0 + 16 + r) * Cout + nb;
        d0[l]      = acc00[r];
        d0[16 + l] = acc01[r];
        d1[l]      = acc10[r];
        d1[16 + l] = acc11[r];
    }
}

// ---------------- batchnorm statistics: one block per channel --------------
__global__ void bn_stats_kernel(const float* __restrict__ x, int P, int C,
                                float* __restrict__ mean, float* __restrict__ var)
{
    __shared__ float s1[TPB], s2[TPB];
    const int c = blockIdx.x, tid = threadIdx.x;
    float a = 0.f, q = 0.f;
    for (int p = tid; p < P; p += TPB) {
        float v = x[(size_t)p * C + c];
        a += v; q += v * v;
    }
    s1[tid] = a; s2[tid] = q;
    __syncthreads();
    for (int s = TPB / 2; s > 0; s >>= 1) {
        if (tid < s) { s1[tid] += s1[tid + s]; s2[tid] += s2[tid + s]; }
        __syncthreads();
    }
    if (tid == 0) {
        float mu = s1[0] / (float)P;
        mean[c] = mu;
        var[c]  = s2[0] / (float)P - mu * mu;   // biased variance (jnp.var)
    }
}

__global__ void bn_relu_kernel(float* __restrict__ x, const float* __restrict__ mean,
                               const float* __restrict__ var, const float* __restrict__ g,
                               const float* __restrict__ bt, long total, int C)
{
    long t = (long)blockIdx.x * blockDim.x + threadIdx.x;
    if (t >= total) return;
    int c = (int)(t % C);
    float v = x[t];
    v = g[c] * (v - mean[c]) * rsqrtf(var[c] + 1e-5f) + bt[c];
    x[t] = fmaxf(v, 0.0f);
}

// ---------------- max pool over K, write into concat slot ------------------
__global__ void maxpool_kernel(const float* __restrict__ feat, int M, int K, int C,
                               float* __restrict__ out, int Ctot, int coff)
{
    long t = (long)blockIdx.x * blockDim.x + threadIdx.x;
    long total = (long)BATCH * M * C;
    if (t >= total) return;
    int  c = (int)(t % C);
    long q = t / C;            // b*M + m
    const float* f = feat + (q * K) * (long)C + c;
    float mx = f[0];
    for (int k = 1; k < K; ++k) mx = fmaxf(mx, f[(long)k * C]);
    out[q * Ctot + coff + c] = mx;
}

// ---------------- 3-NN (smallest distances, first-index ties) --------------
__global__ void knn3_kernel(const float* __restrict__ xyz1, const float* __restrict__ xyz2,
                            int N1, int N2, int* __restrict__ idx3, float* __restrict__ w3)
{
    int t = blockIdx.x * blockDim.x + threadIdx.x;
    if (t >= BATCH * N1) return;
    int b = t / N1;
    const float qx = xyz1[t * 3 + 0], qy = xyz1[t * 3 + 1], qz = xyz1[t * 3 + 2];
    const float* x2 = xyz2 + (size_t)b * N2 * 3;
    float d0 = 1e30f, d1 = 1e30f, d2 = 1e30f;
    int   i0 = 0, i1 = 0, i2 = 0;
    for (int n = 0; n < N2; ++n) {
        float dx = x2[n * 3 + 0] - qx;
        float dy = x2[n * 3 + 1] - qy;
        float dz = x2[n * 3 + 2] - qz;
        float d  = dx * dx + dy * dy + dz * dz;
        if      (d < d0) { d2 = d1; i2 = i1; d1 = d0; i1 = i0; d0 = d; i0 = n; }
        else if (d < d1) { d2 = d1; i2 = i1; d1 = d;  i1 = n; }
        else if (d < d2) { d2 = d;  i2 = n; }
    }
    float w0 = 1.f / (d0 + 1e-8f), w1 = 1.f / (d1 + 1e-8f), w2 = 1.f / (d2 + 1e-8f);
    float s  = w0 + w1 + w2;
    idx3[t * 3 + 0] = i0; idx3[t * 3 + 1] = i1; idx3[t * 3 + 2] = i2;
    w3[t * 3 + 0] = w0 / s; w3[t * 3 + 1] = w1 / s; w3[t * 3 + 2] = w2 / s;
}

// -------- interpolate + concat into K-padded MLP input rows ----------------
__global__ void interp_concat_kernel(const float* __restrict__ p2, int C2,
                                     const float* __restrict__ p1a, int C1a,
                                     const float* __restrict__ p1b, int C1b,
                                     const int* __restrict__ idx3, const float* __restrict__ w3,
                                     int N1, int N2, int Ctp, float* __restrict__ out)
{
    const int Ct = C2 + C1a + C1b;
    long t = (long)blockIdx.x * blockDim.x + threadIdx.x;
    long total = (long)BATCH * N1 * Ctp;
    if (t >= total) return;
    int  c = (int)(t % Ctp);
    long q = t / Ctp;          // b*N1 + n1
    long b = q / N1;
    float v = 0.0f;
    if (c < C2) {
        const int*   id = idx3 + q * 3;
        const float* w  = w3 + q * 3;
        const float* base = p2 + b * (long)N2 * C2;
        v = w[0] * base[(long)id[0] * C2 + c]
          + w[1] * base[(long)id[1] * C2 + c]
          + w[2] * base[(long)id[2] * C2 + c];
    } else if (c < C2 + C1a) {
        v = p1a[q * (long)C1a + (c - C2)];
    } else if (c < Ct) {
        v = p1b[q * (long)C1b + (c - C2 - C1a)];
    }
    out[t] = v;
}

// ---------------- classifier (Cout=2) + transpose to [B, NCLS, N] ----------
__global__ void cls_kernel(const float* __restrict__ x, const float* __restrict__ W,
                           const float* __restrict__ bias, int Np, int Cin, int O,
                           float* __restrict__ out)
{
    long t = (long)blockIdx.x * blockDim.x + threadIdx.x;
    long total = (long)BATCH * Np * O;
    if (t >= total) return;
    int  o = (int)(t % O);
    long q = t / O;            // b*Np + n
    long b = q / Np;
    long n = q % Np;
    const float* xr = x + q * (long)Cin;
    const float* wr = W + (long)o * Cin;
    float s = bias[o];
    for (int c = 0; c < Cin; ++c) s += xr[c] * wr[c];
    out[(b * O + o) * (long)Np + n] = s;
}

__global__ void copy_kernel(const float* __restrict__ s, float* __restrict__ d, long n)
{
    long t = (long)blockIdx.x * blockDim.x + threadIdx.x;
    if (t < n) d[t] = s[t];
}

// ===========================================================================
extern "C" void kernel_launch(void* const* d_in, const int* in_sizes, int n_in,
                              void* d_out, int out_size, void* d_ws, size_t ws_size,
                              hipStream_t stream)
{
    if (n_in < 85) return;   // unexpected flattening; bail deterministically

    auto F = [&](int i) { return (const float*)d_in[i]; };

    // Flattened in setup_inputs() dict insertion order:
    // l0_xyz, l0_points, sa1(b0 L0 w,g,b ...), sa2, sa3, fp1, fp2, fp3,
    // conv1_w, bn1_g, bn1_b, conv2_w, bn2_g, bn2_b, cls_w, cls_b
    const float* l0_xyz    = F(0);
    const float* l0_points = F(1);

    struct Layer { const float* w; const float* g; const float* b; int cin, cinp, cout; };
    auto mkL = [&](int base, int cin, int cout) {
        Layer L; L.w = F(base); L.g = F(base + 1); L.b = F(base + 2);
        L.cin = cin; L.cinp = cin; L.cout = cout; return L;
    };
    Layer sa1b0[3] = { mkL( 2,   5, 32), mkL( 5, 32, 32), mkL( 8, 32,  64) };
    Layer sa1b1[3] = { mkL(11,   5, 64), mkL(14, 64, 64), mkL(17, 64, 128) };
    Layer sa2b0[3] = { mkL(20, 195, 32), mkL(23, 32, 32), mkL(26, 32,  64) };
    Layer sa2b1[3] = { mkL(29, 195, 64), mkL(32, 64, 64), mkL(35, 64, 128) };
    Layer sa3b0[3] = { mkL(38, 195, 64), mkL(41, 64, 64), mkL(44, 64, 128) };
    Layer sa3b1[3] = { mkL(47, 195, 64), mkL(50, 64, 64), mkL(53, 64, 128) };
    Layer fp1L[2]  = { mkL(56, 448, 256), mkL(59, 256, 256) };
    Layer fp2L[2]  = { mkL(62, 448, 128), mkL(65, 128, 128) };
    Layer fp3L[3]  = { mkL(68, 133, 128), mkL(71, 128, 128), mkL(74, 128, 128) };
    const float* conv1_w = F(77); const float* bn1_g = F(78); const float* bn1_b = F(79);
    const float* conv2_w = F(80); const float* bn2_g = F(81); const float* bn2_b = F(82);
    const float* cls_w   = F(83); const float* cls_b = F(84);

    // ---------------- workspace carve-up (floats; every chunk %4==0) -------
    float* ws = (float*)d_ws;
    size_t off = 0;
    auto alloc = [&](size_t n) { float* p = ws + off; off += (n + 3) & ~(size_t)3; return p; };
    int*   fps_idx = (int*)alloc(8 * 1024);
    float* l1_xyz  = alloc(8 * 1024 * 3);
    float* l2_xyz  = alloc(8 * 512 * 3);
    float* l3_xyz  = alloc(8 * 256 * 3);
    float* l1_pts  = alloc((size_t)8 * 1024 * 192);
    float* l2_pts  = alloc((size_t)8 * 512 * 192);
    float* l3_pts  = alloc((size_t)8 * 256 * 256);
    float* l2_fp   = alloc((size_t)8 * 512 * 256);
    float* l1_fp   = alloc((size_t)8 * 1024 * 128);
    float* l0_fe   = alloc((size_t)8 * 4096 * 128);
    int*   gi      = (int*)alloc((size_t)8 * 1024 * 32);
    int*   idx3    = (int*)alloc((size_t)8 * 4096 * 3);
    float* w3      = alloc((size_t)8 * 4096 * 3);
    float* mean_   = alloc(256);
    float* var_    = alloc(256);
    float* featA   = alloc((size_t)262144 * 128);
    float* featB   = alloc((size_t)262144 * 128);
    (void)ws_size; (void)in_sizes; (void)out_size;

    auto nblk = [](long total, int tpb) { return (unsigned)((total + tpb - 1) / tpb); };

    // Zero-pad first-layer weights whose Cin % 4 != 0 into workspace copies.
    auto pad_weight = [&](Layer& L) {
        int cinp = (L.cin + 3) & ~3;
        if (cinp == L.cin) return;
        float* dst = alloc((size_t)L.cout * cinp);
        long tot = (long)L.cout * cinp;
        pad_w_kernel<<<nblk(tot, TPB), TPB, 0, stream>>>(L.w, dst, L.cout, L.cin, cinp);
        L.w = dst; L.cinp = cinp;
    };
    pad_weight(sa1b0[0]); pad_weight(sa1b1[0]);
    pad_weight(sa2b0[0]); pad_weight(sa2b1[0]);
    pad_weight(sa3b0[0]); pad_weight(sa3b1[0]);
    pad_weight(fp3L[0]);

    // GEMM + BN(stats) + ReLU chain, ping-ponging in/out buffers.
    auto run_mlp = [&](float* bin, float* bout, int P, const Layer* L, int nL) -> float* {
        float* in = bin; float* out = bout;
        for (int i = 0; i < nL; ++i) {
            dim3 g(P / 128, L[i].cout / 32);
            wmma_gemm_kernel<<<g, 128, 0, stream>>>(in, L[i].w, out, P, L[i].cinp, L[i].cout);
            bn_stats_kernel<<<L[i].cout, TPB, 0, stream>>>(out, P, L[i].cout, mean_, var_);
            long tot = (long)P * L[i].cout;
            bn_relu_kernel<<<nblk(tot, TPB), TPB, 0, stream>>>(out, mean_, var_, L[i].g, L[i].b, tot, L[i].cout);
            float* t = in; in = out; out = t;
        }
        return in;
    };

    auto run_sa = [&](const float* xyz, const float* pts, int N, int Cp, int M,
                      float* nxyz, float* opts, int Ctot,
                      const float* radii, const int* Ks, Layer* const* branches) {
        fps_kernel<<<BATCH, 1024, 0, stream>>>(xyz, N, M, fps_idx);
        gather_xyz_kernel<<<nblk((long)BATCH * M * 3, TPB), TPB, 0, stream>>>(xyz, fps_idx, N, M, nxyz);
        int coff = 0;
        for (int j = 0; j < 2; ++j) {
            int K = Ks[j]; float r2 = radii[j] * radii[j];
            ball_query_kernel<<<nblk((long)BATCH * M, TPB), TPB, 0, stream>>>(xyz, nxyz, N, M, r2, K, gi);
            int Ctp = branches[j][0].cinp;           // padded input row stride
            long gt = (long)BATCH * M * K * Ctp;
            group_kernel<<<nblk(gt, TPB), TPB, 0, stream>>>(xyz, pts, nxyz, gi, N, M, K, Cp, Ctp, featA);
            float* res = run_mlp(featA, featB, BATCH * M * K, branches[j], 3);
            int Cl = branches[j][2].cout;
            maxpool_kernel<<<nblk((long)BATCH * M * Cl, TPB), TPB, 0, stream>>>(res, M, K, Cl, opts, Ctot, coff);
            coff += Cl;
        }
    };

    auto run_fp = [&](const float* xyz1, int N1, const float* xyz2, int N2,
                      const float* p1a, int C1a, const float* p1b, int C1b,
                      const float* p2, int C2, const Layer* L, int nL, float* outbuf) {
        knn3_kernel<<<nblk((long)BATCH * N1, TPB), TPB, 0, stream>>>(xyz1, xyz2, N1, N2, idx3, w3);
        int Ctp = L[0].cinp;
        long tot = (long)BATCH * N1 * Ctp;
        interp_concat_kernel<<<nblk(tot, TPB), TPB, 0, stream>>>(p2, C2, p1a, C1a, p1b, C1b,
                                                                 idx3, w3, N1, N2, Ctp, featA);
        float* res = run_mlp(featA, featB, BATCH * N1, L, nL);
        long n = (long)BATCH * N1 * L[nL - 1].cout;
        copy_kernel<<<nblk(n, TPB), TPB, 0, stream>>>(res, outbuf, n);
    };

    // ---------------- pipeline ----------------
    { float radii[2] = {1.f, 3.f}; int Ks[2] = {8, 32};
      Layer* br[2] = {sa1b0, sa1b1};
      run_sa(l0_xyz, l0_points, 4096, 2, 1024, l1_xyz, l1_pts, 192, radii, Ks, br); }
    { float radii[2] = {2.f, 4.f}; int Ks[2] = {8, 32};
      Layer* br[2] = {sa2b0, sa2b1};
      run_sa(l1_xyz, l1_pts, 1024, 192, 512, l2_xyz, l2_pts, 192, radii, Ks, br); }
    { float radii[2] = {3.f, 6.f}; int Ks[2] = {16, 32};
      Layer* br[2] = {sa3b0, sa3b1};
      run_sa(l2_xyz, l2_pts, 512, 192, 256, l3_xyz, l3_pts, 256, radii, Ks, br); }

    run_fp(l2_xyz,  512, l3_xyz,  256, l2_pts,    192, nullptr, 0, l3_pts, 256, fp1L, 2, l2_fp);
    run_fp(l1_xyz, 1024, l2_xyz,  512, l1_pts,    192, nullptr, 0, l2_fp,  256, fp2L, 2, l1_fp);
    run_fp(l0_xyz, 4096, l1_xyz, 1024, l0_points,   2, l0_xyz,  3, l1_fp,  128, fp3L, 3, l0_fe);

    // head: conv1(128->256)+BN+ReLU, conv2(256->128)+BN+ReLU, classifier
    Layer headL[2] = { {conv1_w, bn1_g, bn1_b, 128, 128, 256},
                       {conv2_w, bn2_g, bn2_b, 256, 256, 128} };
    float* hres = run_mlp(l0_fe, featA, BATCH * 4096, headL, 2);
    cls_kernel<<<nblk((long)BATCH * 4096 * 2, TPB), TPB, 0, stream>>>(
        hres, cls_w, cls_b, 4096, 128, 2, (float*)d_out);
}